// PureQuantumWalkEngine_20976620274068
// MI455X (gfx1250) — compile-verified
//
#include <hip/hip_runtime.h>
#include <math.h>

typedef float v2f __attribute__((ext_vector_type(2)));
typedef float v8f __attribute__((ext_vector_type(8)));

#define NCELLS 4096
#define HDIM   2048
#define INDIM  1024
#define OUTDIM 1024
#define HCN    4096
#define HCBITS 12
#define KCOMB  3072      // IN + H
#define KMEM   1025      // OUT + 1
#define KMEMP  1028      // padded stride

// ---------------------------------------------------------------- reductions
__device__ inline float block_allreduce_sum(float v) {
  __shared__ float sm[9];
  int lane = threadIdx.x & 31, wv = threadIdx.x >> 5;
  #pragma unroll
  for (int m = 16; m >= 1; m >>= 1) v += __shfl_xor(v, m, 32);
  __syncthreads();
  if (lane == 0) sm[wv] = v;
  __syncthreads();
  if (threadIdx.x == 0) {
    float r = 0.0f; int nw = blockDim.x >> 5;
    for (int i = 0; i < nw; ++i) r += sm[i];
    sm[8] = r;
  }
  __syncthreads();
  return sm[8];
}

__device__ inline float block_allreduce_max(float v) {
  __shared__ float sm[9];
  int lane = threadIdx.x & 31, wv = threadIdx.x >> 5;
  #pragma unroll
  for (int m = 16; m >= 1; m >>= 1) v = fmaxf(v, __shfl_xor(v, m, 32));
  __syncthreads();
  if (lane == 0) sm[wv] = v;
  __syncthreads();
  if (threadIdx.x == 0) {
    float r = -3.4e38f; int nw = blockDim.x >> 5;
    for (int i = 0; i < nw; ++i) r = fmaxf(r, sm[i]);
    sm[8] = r;
  }
  __syncthreads();
  return sm[8];
}

// ---------------------------------------------------------------- WMMA core
__device__ inline v8f wmma_f32(v2f a, v2f b, v8f c) {
  return __builtin_amdgcn_wmma_f32_16x16x4_f32(
      false, a, false, b, (short)0, c, false, false);
}

// Unguarded fragment load (caller guarantees in-bounds; K multiple of 4).
// A layout: lanes 0-15: M=lane, {K=k0,k0+1}; lanes 16-31: M=lane-16, {K=k0+2,k0+3}.
// B mirrors with N = lane%16.
__device__ inline v2f load_frag(const float* __restrict__ P, int ld,
                                int row0, int k0, int lane) {
  const float* p = P + (size_t)(row0 + (lane & 15)) * ld + (k0 + 2 * (lane >> 4));
  v2f v;
  v.x = p[0];
  v.y = p[1];
  return v;
}

// Branchless bounds-guarded load: clamp address (always legal), select value.
// Keeps EXEC all-ones (required for WMMA) and avoids exec-masked branches.
__device__ inline v2f load_frag_clamp(const float* __restrict__ P, int ld,
                                      int row0, int k0, int K, int lane) {
  int r = row0 + (lane & 15);
  int k = k0 + 2 * (lane >> 4);
  int kc0 = (k     < K) ? k     : (K - 1);
  int kc1 = (k + 1 < K) ? k + 1 : (K - 1);
  const float* base = P + (size_t)r * ld;
  float x = base[kc0];
  float y = base[kc1];
  v2f v;
  v.x = (k     < K) ? x : 0.0f;
  v.y = (k + 1 < K) ? y : 0.0f;
  return v;
}

__device__ inline void zero_acc(v8f& a) {
  #pragma unroll
  for (int i = 0; i < 8; ++i) a[i] = 0.0f;
}

// ---------------------------------------------------------------- walk stage
__global__ __launch_bounds__(256) void zero_f32(float* p, int n) {
  int i = blockIdx.x * 256 + threadIdx.x;
  if (i < n) p[i] = 0.0f;
}

__global__ __launch_bounds__(256) void walk_coin(
    const float* __restrict__ ar, const float* __restrict__ ai,
    const float* __restrict__ cr, const float* __restrict__ ci,
    float* __restrict__ c1r, float* __restrict__ c1i, float* __restrict__ s0) {
  int n = blockIdx.x;
  float cr00 = cr[0], cr01 = cr[1], cr10 = cr[2], cr11 = cr[3];
  float ci00 = ci[0], ci01 = ci[1], ci10 = ci[2], ci11 = ci[3];
  const float* a0r = ar + (size_t)n * 2 * HDIM;
  const float* a1r = a0r + HDIM;
  const float* a0i = ai + (size_t)n * 2 * HDIM;
  const float* a1i = a0i + HDIM;
  float acc = 0.0f;
  for (int d = threadIdx.x; d < HDIM; d += 256) {
    float x0r = a0r[d], x1r = a1r[d], x0i = a0i[d], x1i = a1i[d];
    float r0 = cr00 * x0r - ci00 * x0i + cr01 * x1r - ci01 * x1i;
    float i0 = cr00 * x0i + ci00 * x0r + cr01 * x1i + ci01 * x1r;
    float r1 = cr10 * x0r - ci10 * x0i + cr11 * x1r - ci11 * x1i;
    float i1 = cr10 * x0i + ci10 * x0r + cr11 * x1i + ci11 * x1r;
    c1r[(size_t)n * HDIM + d] = r1;
    c1i[(size_t)n * HDIM + d] = i1;
    acc += r0 * r0 + i0 * i0;
  }
  float tot = block_allreduce_sum(acc);
  if (threadIdx.x == 0) s0[n] = tot;
}

__global__ __launch_bounds__(256) void walk_gather(
    const float* __restrict__ c1r, const float* __restrict__ c1i,
    float* __restrict__ s1) {
  int n = blockIdx.x;
  const float inv = 1.0f / (float)HCBITS;
  float acc = 0.0f;
  for (int d = threadIdx.x; d < HDIM; d += 256) {
    float sr = 0.0f, si = 0.0f;
    #pragma unroll
    for (int bit = 0; bit < HCBITS; ++bit) {
      int nb = n ^ (1 << bit);
      sr += c1r[(size_t)nb * HDIM + d];
      si += c1i[(size_t)nb * HDIM + d];
    }
    sr *= inv; si *= inv;
    acc += sr * sr + si * si;
  }
  float tot = block_allreduce_sum(acc);
  if (threadIdx.x == 0) s1[n] = tot;
}

__global__ __launch_bounds__(256) void walk_probs(
    const float* __restrict__ s0, const float* __restrict__ s1,
    float* __restrict__ probs) {
  float loc = 0.0f;
  for (int n = threadIdx.x; n < HCN; n += 256) loc += s0[n] + s1[n];
  float total = block_allreduce_sum(loc);
  float norm = sqrtf(total);
  float inv2 = 1.0f / ((norm + 1e-8f) * (norm + 1e-8f));
  float psum = total * inv2;
  float scale = inv2 / (psum + 1e-8f);
  for (int n = threadIdx.x; n < HCN; n += 256) probs[n] = (s0[n] + s1[n]) * scale;
}

__global__ __launch_bounds__(256) void walk_feat_k(
    const float* __restrict__ probs, const float* __restrict__ Ww,
    const float* __restrict__ bw, float* __restrict__ wf) {
  int lane = threadIdx.x & 31;
  int j = blockIdx.x * 8 + (threadIdx.x >> 5);
  const float* row = Ww + (size_t)j * HCN;
  float acc = 0.0f;
  for (int n = lane; n < HCN; n += 32) acc += probs[n] * row[n];
  #pragma unroll
  for (int m = 16; m >= 1; m >>= 1) acc += __shfl_xor(acc, m, 32);
  if (lane == 0) wf[j] = acc + bw[j];
}

// ------------------------------------------------------- hiddens / combined
__global__ __launch_bounds__(256) void fill_x(const float* __restrict__ x,
                                              float* __restrict__ comb) {
  int idx = blockIdx.x * 256 + threadIdx.x;           // 4096*1024
  int n = idx >> 10, k = idx & 1023;
  comb[(size_t)n * KCOMB + k] = x[k];
}

__global__ __launch_bounds__(256) void upd_hidden(
    const float* __restrict__ hid, const float* __restrict__ probs,
    const float* __restrict__ wf, float* __restrict__ h1,
    float* __restrict__ comb) {
  int idx = blockIdx.x * 256 + threadIdx.x;           // 4096*2048
  int n = idx >> 11, d = idx & 2047;
  float v = hid[idx] * (0.9f + 0.2f * probs[n]) + 0.05f * wf[d];
  h1[idx] = v;
  comb[(size_t)n * KCOMB + INDIM + d] = v;
}

// -------------------------------------------------- generic WMMA f32 GEMM
// C = act(A[M,K] * B[N,K]^T + bias).  8 waves arranged WMxWN; wave tile
// 32(M) x 64(N): 2 A-frags x 4 B-frags -> 8 WMMA per K-step.  Grids are
// exact (no tile guards); K multiple of 4; all loads unguarded & branchless.
template <int WM, int WN>
__global__ __launch_bounds__(256) void gemm_bias(
    const float* __restrict__ A, int lda, const float* __restrict__ B, int ldb,
    const float* __restrict__ bias, float* __restrict__ C, int ldc,
    int K, int relu) {
  int lane = threadIdx.x & 31;
  int wave = threadIdx.x >> 5;
  int m0 = blockIdx.x * (WM * 32) + (wave / WN) * 32;
  int nb = blockIdx.y * (WN * 64) + (wave % WN) * 64;
  v8f acc[2][4];
  #pragma unroll
  for (int s = 0; s < 2; ++s)
    #pragma unroll
    for (int t = 0; t < 4; ++t) zero_acc(acc[s][t]);
  for (int k0 = 0; k0 < K; k0 += 4) {
    v2f a0 = load_frag(A, lda, m0, k0, lane);
    v2f a1 = load_frag(A, lda, m0 + 16, k0, lane);
    #pragma unroll
    for (int t = 0; t < 4; ++t) {
      v2f b = load_frag(B, ldb, nb + t * 16, k0, lane);
      acc[0][t] = wmma_f32(a0, b, acc[0][t]);
      acc[1][t] = wmma_f32(a1, b, acc[1][t]);
    }
  }
  int nc = lane & 15;
  #pragma unroll
  for (int s = 0; s < 2; ++s) {
    int mb = m0 + s * 16 + 8 * (lane >> 4);
    #pragma unroll
    for (int t = 0; t < 4; ++t) {
      int n = nb + t * 16 + nc;
      float bv = bias[n];
      #pragma unroll
      for (int r = 0; r < 8; ++r) {
        float c = acc[s][t][r] + bv;
        c = relu ? fmaxf(c, 0.0f) : c;
        C[(size_t)(mb + r) * ldc + n] = c;
      }
    }
  }
}

// out = (acta*Wa2^T+ba2) - (actg*Wg2^T+bg2); accumulates row sum-of-squares
// into t[].  M=4096, N=1024, K=128; grid (128,4), block tile 32x256, exact.
__global__ __launch_bounds__(256) void mlp2_diff(
    const float* __restrict__ Aa, const float* __restrict__ Ag,
    const float* __restrict__ Wa, const float* __restrict__ Wg,
    const float* __restrict__ ba, const float* __restrict__ bg,
    float* __restrict__ C, float* __restrict__ t) {
  const int K = 128, N = OUTDIM;
  int lane = threadIdx.x & 31;
  int wave = threadIdx.x >> 5;
  int m0 = blockIdx.x * 32 + (wave >> 2) * 16;
  int nb = blockIdx.y * 256 + (wave & 3) * 64;
  v8f aacc[4], gacc[4];
  #pragma unroll
  for (int tt = 0; tt < 4; ++tt) { zero_acc(aacc[tt]); zero_acc(gacc[tt]); }
  for (int k0 = 0; k0 < K; k0 += 4) {
    v2f fa = load_frag(Aa, K, m0, k0, lane);
    v2f fg = load_frag(Ag, K, m0, k0, lane);
    #pragma unroll
    for (int tt = 0; tt < 4; ++tt) {
      int n0 = nb + tt * 16;
      v2f wa = load_frag(Wa, K, n0, k0, lane);
      v2f wg = load_frag(Wg, K, n0, k0, lane);
      aacc[tt] = wmma_f32(fa, wa, aacc[tt]);
      gacc[tt] = wmma_f32(fg, wg, gacc[tt]);
    }
  }
  int mb = m0 + 8 * (lane >> 4);
  int nc = lane & 15;
  float sq[8];
  #pragma unroll
  for (int r = 0; r < 8; ++r) sq[r] = 0.0f;
  #pragma unroll
  for (int tt = 0; tt < 4; ++tt) {
    int n = nb + tt * 16 + nc;
    float bv = ba[n] - bg[n];
    #pragma unroll
    for (int r = 0; r < 8; ++r) {
      float c = aacc[tt][r] - gacc[tt][r] + bv;
      C[(size_t)(mb + r) * N + n] = c;
      sq[r] += c * c;
    }
  }
  #pragma unroll
  for (int r = 0; r < 8; ++r) {
    float v = sq[r];
    v += __shfl_xor(v, 1, 32); v += __shfl_xor(v, 2, 32);
    v += __shfl_xor(v, 4, 32); v += __shfl_xor(v, 8, 32);
    if ((lane & 15) == 0) atomicAdd(&t[mb + r], v);
  }
}

__global__ __launch_bounds__(256) void t_norm(float* t) {
  int i = blockIdx.x * 256 + threadIdx.x;
  if (i < NCELLS) t[i] *= (1.0f / (float)OUTDIM);
}

__global__ __launch_bounds__(256) void memin_fill(
    const float* __restrict__ outb, const float* __restrict__ t,
    float* __restrict__ mi) {
  int idx = blockIdx.x * 256 + threadIdx.x;           // 4096*1028
  int n = idx / KMEMP, k = idx % KMEMP;
  float v = 0.0f;
  if (k < OUTDIM)       v = outb[(size_t)n * OUTDIM + k];
  else if (k == OUTDIM) v = t[n];
  mi[idx] = v;
}

// -------------------------------------------------- fused GRU (WMMA)
// One wave computes a 16(M) x 16(H) tile of new_h: 6 WMMA accumulators
// (r/z/n x {ih,hh}); gi/gh never materialized.  W_ih K=1025 handled as an
// unguarded main loop over 1024 plus one clamped (branchless) tail step.
__global__ __launch_bounds__(256) void gru_fused(
    const float* __restrict__ mi, const float* __restrict__ Wih,
    const float* __restrict__ bih, const float* __restrict__ h1,
    const float* __restrict__ Whh, const float* __restrict__ bhh,
    float* __restrict__ nh) {
  int lane = threadIdx.x & 31;
  int wave = threadIdx.x >> 5;
  int tile = blockIdx.x * 8 + wave;
  const int tiles_j = HDIM / 16;                      // 128
  int m0 = (tile / tiles_j) * 16;
  int j0 = (tile % tiles_j) * 16;
  v8f ai[3], ah[3];
  #pragma unroll
  for (int g = 0; g < 3; ++g) { zero_acc(ai[g]); zero_acc(ah[g]); }
  for (int k0 = 0; k0 < 1024; k0 += 4) {              // unguarded main loop
    v2f a = load_frag(mi, KMEMP, m0, k0, lane);
    #pragma unroll
    for (int g = 0; g < 3; ++g) {
      v2f b = load_frag(Wih, KMEM, g * HDIM + j0, k0, lane);
      ai[g] = wmma_f32(a, b, ai[g]);
    }
  }
  {                                                   // tail: k0 = 1024
    v2f a = load_frag(mi, KMEMP, m0, 1024, lane);     // mem_in zero-padded
    #pragma unroll
    for (int g = 0; g < 3; ++g) {
      v2f b = load_frag_clamp(Wih, KMEM, g * HDIM + j0, 1024, KMEM, lane);
      ai[g] = wmma_f32(a, b, ai[g]);
    }
  }
  for (int k0 = 0; k0 < HDIM; k0 += 4) {
    v2f a = load_frag(h1, HDIM, m0, k0, lane);
    #pragma unroll
    for (int g = 0; g < 3; ++g) {
      v2f b = load_frag(Whh, HDIM, g * HDIM + j0, k0, lane);
      ah[g] = wmma_f32(a, b, ah[g]);
    }
  }
  int mb = m0 + 8 * (lane >> 4);
  int j = j0 + (lane & 15);
  float bri = bih[j], bzi = bih[HDIM + j], bni = bih[2 * HDIM + j];
  float brh = bhh[j], bzh = bhh[HDIM + j], bnh = bhh[2 * HDIM + j];
  #pragma unroll
  for (int r = 0; r < 8; ++r) {
    int m = mb + r;
    float hij = h1[(size_t)m * HDIM + j];
    float rr = 1.0f / (1.0f + expf(-(ai[0][r] + bri + ah[0][r] + brh)));
    float zz = 1.0f / (1.0f + expf(-(ai[1][r] + bzi + ah[1][r] + bzh)));
    float nn = tanhf(ai[2][r] + bni + rr * (ah[2][r] + bnh));
    nh[(size_t)m * HDIM + j] = (1.0f - zz) * nn + zz * hij;
  }
}

// -------------------------------------------------- faction sync
__global__ __launch_bounds__(256) void fmean_k(const float* __restrict__ nh,
                                               float* __restrict__ fm) {
  int f = blockIdx.x;
  int d = blockIdx.y * 256 + threadIdx.x;
  float s = 0.0f;
  for (int row = 0; row < 512; ++row)
    s += nh[(size_t)((f << 9) + row) * HDIM + d];
  fm[f * HDIM + d] = s * (1.0f / 512.0f);
}

__global__ __launch_bounds__(256) void gmean_k(const float* __restrict__ fm,
                                               float* __restrict__ gm) {
  int d = blockIdx.x * 256 + threadIdx.x;
  float s = 0.0f;
  #pragma unroll
  for (int f = 0; f < 8; ++f) s += fm[f * HDIM + d];
  gm[d] = s * 0.125f;   // per-faction mean is invariant under the sync shrink
}

__global__ __launch_bounds__(256) void faction_apply(
    const float* __restrict__ nh, const float* __restrict__ fm,
    const float* __restrict__ gm, const int* __restrict__ step,
    float* __restrict__ dst) {
  int idx = blockIdx.x * 256 + threadIdx.x;           // 4096*2048
  int n = idx >> 11, d = idx & 2047;
  int f = n >> 9, rin = n & 511;
  float v = 0.85f * nh[idx] + 0.15f * fm[f * HDIM + d];
  if (*step > 5 && rin < 128) v = 0.85f * v + 0.15f * gm[d];
  dst[idx] = v;
}

// -------------------------------------------------- output head
__global__ __launch_bounds__(256) void softmax_t(
    const float* __restrict__ t, float* __restrict__ w,
    float* __restrict__ tmean_out) {
  float lmax = -3.4e38f;
  for (int n = threadIdx.x; n < NCELLS; n += 256) lmax = fmaxf(lmax, t[n]);
  float gmax = block_allreduce_max(lmax);
  float lsum = 0.0f, lts = 0.0f;
  for (int n = threadIdx.x; n < NCELLS; n += 256) {
    lsum += expf(t[n] - gmax);
    lts += t[n];
  }
  float gsum = block_allreduce_sum(lsum);
  float gts = block_allreduce_sum(lts);
  float inv = 1.0f / gsum;
  for (int n = threadIdx.x; n < NCELLS; n += 256) w[n] = expf(t[n] - gmax) * inv;
  if (threadIdx.x == 0) tmean_out[0] = gts * (1.0f / (float)NCELLS);
}

__global__ __launch_bounds__(256) void comb_k(const float* __restrict__ w,
                                              const float* __restrict__ outb,
                                              float* __restrict__ cv) {
  int j = blockIdx.x * 256 + threadIdx.x;             // 1024
  float acc = 0.0f;
  for (int n = 0; n < NCELLS; ++n) acc += w[n] * outb[(size_t)n * OUTDIM + j];
  cv[j] = acc;
}

__global__ __launch_bounds__(256) void pred_k(const float* __restrict__ cv,
                                              const float* __restrict__ Wo,
                                              const float* __restrict__ bo,
                                              float* __restrict__ pred) {
  int lane = threadIdx.x & 31;
  int i = blockIdx.x * 8 + (threadIdx.x >> 5);        // 1024 outputs
  const float* row = Wo + (size_t)i * OUTDIM;
  float acc = 0.0f;
  for (int j = lane; j < OUTDIM; j += 32) acc += cv[j] * row[j];
  #pragma unroll
  for (int m = 16; m >= 1; m >>= 1) acc += __shfl_xor(acc, m, 32);
  if (lane == 0) pred[i] = acc + bo[i];
}

// ==================================================================== launch
extern "C" void kernel_launch(void* const* d_in, const int* in_sizes, int n_in,
                              void* d_out, int out_size, void* d_ws, size_t ws_size,
                              hipStream_t stream) {
  const float* x     = (const float*)d_in[0];
  const float* hid   = (const float*)d_in[1];
  const float* ampr  = (const float*)d_in[2];
  const float* ampi  = (const float*)d_in[3];
  const float* coinr = (const float*)d_in[4];
  const float* coini = (const float*)d_in[5];
  const float* Ww    = (const float*)d_in[6];
  const float* bw    = (const float*)d_in[7];
  const float* Wa1   = (const float*)d_in[8];
  const float* ba1   = (const float*)d_in[9];
  const float* Wa2   = (const float*)d_in[10];
  const float* ba2   = (const float*)d_in[11];
  const float* Wg1   = (const float*)d_in[12];
  const float* bg1   = (const float*)d_in[13];
  const float* Wg2   = (const float*)d_in[14];
  const float* bg2   = (const float*)d_in[15];
  const float* Wih   = (const float*)d_in[16];
  const float* bih   = (const float*)d_in[17];
  const float* Whh   = (const float*)d_in[18];
  const float* bhh   = (const float*)d_in[19];
  const float* Wo    = (const float*)d_in[20];
  const float* bo    = (const float*)d_in[21];
  const int*   step  = (const int*)d_in[22];

  float* out  = (float*)d_out;   // [0..1023] pred, [1024] t.mean, rest new_hiddens
  float* pred = out;
  float* tmn  = out + 1024;
  float* nhid = out + 1025;

  const size_t NM = (size_t)NCELLS * HDIM;            // 8388608
  float* ws    = (float*)d_ws;
  float* c1r   = ws;
  float* c1i   = c1r + NM;
  float* h1    = c1i + NM;
  float* comb  = h1 + NM;                             // 4096*3072
  float* newh  = comb + (size_t)NCELLS * KCOMB;
  float* memin = newh + NM;                           // 4096*1028
  float* outb  = memin + (size_t)NCELLS * KMEMP;      // 4096*1024
  float* acta  = outb + (size_t)NCELLS * OUTDIM;      // 4096*128
  float* actg  = acta + (size_t)NCELLS * 128;
  float* s0    = actg + (size_t)NCELLS * 128;
  float* s1    = s0 + NCELLS;
  float* probs = s1 + NCELLS;
  float* tt    = probs + NCELLS;
  float* wgt   = tt + NCELLS;
  float* wf    = wgt + NCELLS;
  float* fm    = wf + HDIM;
  float* gm    = fm + 8 * HDIM;
  float* cv    = gm + HDIM;                           // 1024

  // --- quantum walk (phase drops out of |.|^2; only probs survive) ---
  zero_f32<<<16, 256, 0, stream>>>(tt, NCELLS);
  walk_coin<<<HCN, 256, 0, stream>>>(ampr, ampi, coinr, coini, c1r, c1i, s0);
  walk_gather<<<HCN, 256, 0, stream>>>(c1r, c1i, s1);
  walk_probs<<<1, 256, 0, stream>>>(s0, s1, probs);
  walk_feat_k<<<HDIM / 8, 256, 0, stream>>>(probs, Ww, bw, wf);

  // --- hidden update + combined assembly ---
  fill_x<<<(NCELLS * INDIM) / 256, 256, 0, stream>>>(x, comb);
  upd_hidden<<<NM / 256, 256, 0, stream>>>(hid, probs, wf, h1, comb);

  // --- dual MLP (WMMA) ---  M=4096, N=128, K=3072: 4x2 waves, block 128x128
  gemm_bias<4, 2><<<dim3(NCELLS / 128, 1), 256, 0, stream>>>(
      comb, KCOMB, Wa1, KCOMB, ba1, acta, 128, KCOMB, 1);
  gemm_bias<4, 2><<<dim3(NCELLS / 128, 1), 256, 0, stream>>>(
      comb, KCOMB, Wg1, KCOMB, bg1, actg, 128, KCOMB, 1);
  mlp2_diff<<<dim3(NCELLS / 32, OUTDIM / 256), 256, 0, stream>>>(
      acta, actg, Wa2, Wg2, ba2, bg2, outb, tt);
  t_norm<<<16, 256, 0, stream>>>(tt);

  // --- fused GRU (WMMA, no gi/gh materialization) ---
  memin_fill<<<(NCELLS * KMEMP) / 256, 256, 0, stream>>>(outb, tt, memin);
  gru_fused<<<(NCELLS / 16) * (HDIM / 16) / 8, 256, 0, stream>>>(
      memin, Wih, bih, h1, Whh, bhh, newh);

  // --- faction sync -> new_hiddens output ---
  fmean_k<<<dim3(8, HDIM / 256), 256, 0, stream>>>(newh, fm);
  gmean_k<<<HDIM / 256, 256, 0, stream>>>(fm, gm);
  faction_apply<<<NM / 256, 256, 0, stream>>>(newh, fm, gm, step, nhid);

  // --- tension-weighted combine + output head ---
  softmax_t<<<1, 256, 0, stream>>>(tt, wgt, tmn);
  comb_k<<<OUTDIM / 256, 256, 0, stream>>>(wgt, outb, cv);
  pred_k<<<OUTDIM / 8, 256, 0, stream>>>(cv, Wo, bo, pred);
}